// Down_wt_2894807957626
// MI455X (gfx1250) — compile-verified
//
#include <hip/hip_runtime.h>

typedef float v2f __attribute__((ext_vector_type(2)));
typedef float v4f __attribute__((ext_vector_type(4)));
typedef float v8f __attribute__((ext_vector_type(8)));

#define IN_CH   64
#define OUT_CH  128
#define HW      256          // input H = W
#define OHW     128          // output H = W
#define CH_STR  (HW * HW)    // channel stride in floats

// ---------------------------------------------------------------------------
// Kernel 1: fold Haar DWT into the 1x1-conv weights, pre-shuffled into WMMA
// A-operand lane order with M-tile pairs interleaved for b128 loads.
//   Wlay element index = ktile*512 + pair*128 + lane*4 + u   (u = 0..3)
//     mtile = pair*2 + (u>>1),  j = u&1
//   A layout (16x4 f32): lane l, vgpr j -> M = l%16, K = 2*(l/16) + j
//   Global K ordering:  K = dy*128 + 2*c + dx   (dy/dx = quadrant row/col)
// ---------------------------------------------------------------------------
__global__ void fold_weights(const float* __restrict__ W, float* __restrict__ Wlay) {
    int idx = blockIdx.x * blockDim.x + threadIdx.x;      // 0 .. 32767
    if (idx >= 32768) return;
    int u     = idx & 3;
    int lane  = (idx >> 2) & 31;
    int pair  = (idx >> 7) & 3;
    int ktile = idx >> 9;
    int mtile = pair * 2 + (u >> 1);
    int j     = u & 1;

    int M = mtile * 16 + (lane & 15);
    int K = ktile * 4 + 2 * (lane >> 4) + j;
    int dy = K >> 7;
    int rr = K & 127;
    int c  = rr >> 1;          // input channel
    int dx = rr & 1;
    int q  = dy * 2 + dx;      // 0=a(0,0) 1=b(0,1) 2=c(1,0) 3=d(1,1)

    float wll = W[M * 256 + 0 * IN_CH + c];
    float wlh = W[M * 256 + 1 * IN_CH + c];
    float whl = W[M * 256 + 2 * IN_CH + c];
    float whh = W[M * 256 + 3 * IN_CH + c];

    float r;
    switch (q) {
        case 0:  r = 0.5f * ( wll - wlh - whl + whh); break;  // a
        case 1:  r = 0.5f * ( wll - wlh + whl - whh); break;  // b
        case 2:  r = 0.5f * ( wll + wlh - whl - whh); break;  // c
        default: r = 0.5f * ( wll + wlh + whl + whh); break;  // d
    }
    Wlay[idx] = r;
}

// ---------------------------------------------------------------------------
// Kernel 2: fused strided-GEMM.  One block (8 wave32) per (batch, out_row);
// each wave: 16 output columns x all 128 output channels (8 acc tiles).
// Per K-step: 1 global_load_b64 (B quad-pixels, fully coalesced per
// half-wave), 4 global_load_b128 (A weights, L0-resident), 8 f32 WMMAs.
// B buffered 3 steps deep, A double-buffered -> loads overlap WMMAs.
// ---------------------------------------------------------------------------
__global__ void __launch_bounds__(256)
dwt_gemm(const float* __restrict__ x, const float* __restrict__ Wlay,
         const float* __restrict__ bias, float* __restrict__ out) {
    const int lane = threadIdx.x & 31;
    const int wave = threadIdx.x >> 5;

    const int b   = blockIdx.x >> 7;      // batch
    const int ho  = blockIdx.x & 127;     // output row
    const int wo0 = wave * 16;            // first output column of this wave

    const int n    = lane & 15;           // B/D column within tile
    const int krow = lane >> 4;           // lane half -> K = 2*krow + j

    v8f zero = {};
    v8f acc[8];
#pragma unroll
    for (int m = 0; m < 8; ++m) acc[m] = zero;

    // B base pointers per quadrant-row dy: x[b, krow, 2*ho+dy, 2*(wo0+n)]
    const float* xb[2];
#pragma unroll
    for (int dy = 0; dy < 2; ++dy)
        xb[dy] = x + (size_t)b * IN_CH * CH_STR
                   + (size_t)krow * CH_STR
                   + (size_t)(2 * ho + dy) * HW
                   + (size_t)(2 * (wo0 + n));

    const float* Wp = Wlay + (lane << 2);

    auto loadB = [&](int s) -> v2f {
        int dy = s >> 5;
        int cb = s & 31;                              // channel = cb*2 + krow
        return *(const v2f*)(xb[dy] + (size_t)(cb * 2) * CH_STR);
    };
    auto loadA = [&](int s, v2f* am) {
#pragma unroll
        for (int pr = 0; pr < 4; ++pr) {
            v4f a4 = *(const v4f*)(Wp + s * 512 + pr * 128);
            am[2 * pr]     = __builtin_shufflevector(a4, a4, 0, 1);
            am[2 * pr + 1] = __builtin_shufflevector(a4, a4, 2, 3);
        }
    };

    v2f bm[4];
    v2f am[2][8];
#pragma unroll
    for (int s = 0; s < 3; ++s) bm[s] = loadB(s);
    loadA(0, am[0]);

#pragma unroll
    for (int s = 0; s < 64; ++s) {
        if (s + 3 < 64) bm[(s + 3) & 3] = loadB(s + 3);
        if (s + 1 < 64) loadA(s + 1, am[(s + 1) & 1]);
        if (s + 8 < 64) {
            int dy = (s + 8) >> 5, cb = (s + 8) & 31;
            __builtin_prefetch(xb[dy] + (size_t)(cb * 2) * CH_STR, 0, 1);
        }
#pragma unroll
        for (int m = 0; m < 8; ++m)
            acc[m] = __builtin_amdgcn_wmma_f32_16x16x4_f32(
                /*neg_a=*/false, am[s & 1][m], /*neg_b=*/false, bm[s & 3],
                /*c_mod=*/(short)0, acc[m],
                /*reuse_a=*/false, /*reuse_b=*/false);
    }

    // Bias + store.  C/D layout: vgpr r, lanes0-15 -> M=r, lanes16-31 -> M=r+8.
    const int o_hi = 8 * krow;
#pragma unroll
    for (int m = 0; m < 8; ++m) {
#pragma unroll
        for (int r = 0; r < 8; ++r) {
            int o = m * 16 + r + o_hi;
            float v = acc[m][r] + bias[o];
            out[(((size_t)b * OUT_CH + o) * OHW + ho) * OHW + (wo0 + n)] = v;
        }
    }
}

// ---------------------------------------------------------------------------
extern "C" void kernel_launch(void* const* d_in, const int* in_sizes, int n_in,
                              void* d_out, int out_size, void* d_ws, size_t ws_size,
                              hipStream_t stream) {
    const float* x    = (const float*)d_in[0];   // [16,64,256,256]
    const float* W    = (const float*)d_in[1];   // [128,256]
    const float* bias = (const float*)d_in[2];   // [128]
    float*       out  = (float*)d_out;           // [16,128,128,128]
    float*       Wlay = (float*)d_ws;            // 32768 floats (128 KB)

    fold_weights<<<128, 256, 0, stream>>>(W, Wlay);
    dwt_gemm<<<16 * OHW, 256, 0, stream>>>(x, Wlay, bias, out);
}